// ReferenceGEMLoss_43911745635095
// MI455X (gfx1250) — compile-verified
//
#include <hip/hip_runtime.h>
#include <stdint.h>

// GEM loss, simplified: per_token = E_q[logits] - logits[label], q = softmax(logits/BETA).
// loss = sum(per_token * valid) / sum(valid).  Single pass over 512MB -> HBM-bound (~22us).

#define WAVES        8
#define DEPTH        4          // async loads in flight per wave
#define SLOTS        8          // LDS ring slots per wave (2*DEPTH: no read/write overlap)
#define IGNORE_INDEX (-100)

// ---------- CDNA5 async global->LDS path (per-wave ASYNCcnt, no barriers) ----------

typedef __attribute__((address_space(3))) unsigned char lds_u8_t;

__device__ __forceinline__ unsigned lds_byte_off(const void* p) {
  // generic -> LDS addrspace cast, ptrtoint gives the 32-bit LDS offset
  return (unsigned)(unsigned long long)(lds_u8_t*)(p);
}

__device__ __forceinline__ void async_copy_b128(const float* sbase, unsigned byte_off,
                                                unsigned lds_off) {
  // GVS mode: 64-bit SGPR base + per-lane 32-bit byte offset; dest = per-lane LDS byte addr.
  asm volatile("global_load_async_to_lds_b128 %0, %1, %2"
               :
               : "v"(lds_off), "v"(byte_off), "s"(sbase)
               : "memory");
}

template <int N>
__device__ __forceinline__ void wait_asynccnt() {
  asm volatile("s_wait_asynccnt %0" :: "n"(N) : "memory");
}

// ---------- online weighted-softmax accumulator (base-2 domain: 1 v_exp_f32/elem) ----------

// state: m = max(u) so far (u = x*log2e/beta), S = sum 2^(u-m), E = sum 2^(u-m)*x
__device__ __forceinline__ void upd(float x, float& m, float& S, float& E) {
  const float K = 2.0609929155556620f;  // log2(e)/0.7
  float u  = x * K;
  float nm = fmaxf(m, u);
  float d  = __builtin_amdgcn_exp2f(fminf(m, u) - nm);  // 2^(smaller - larger)
  bool  up = u > m;
  float a  = up ? 1.0f : d;   // 2^(u - nm)
  float b  = up ? d : 1.0f;   // 2^(m_old - nm)
  S = __builtin_fmaf(S, b, a);
  E = __builtin_fmaf(E, b, a * x);
  m = nm;
}

__device__ __forceinline__ void mrg(float& m, float& S, float& E,
                                    float m2, float S2, float E2) {
  float nm = fmaxf(m, m2);
  float d1 = (m  == nm) ? 1.0f : __builtin_amdgcn_exp2f(m  - nm);  // guards -inf - -inf
  float d2 = (m2 == nm) ? 1.0f : __builtin_amdgcn_exp2f(m2 - nm);
  S = __builtin_fmaf(S, d1, S2 * d2);
  E = __builtin_fmaf(E, d1, E2 * d2);
  m = nm;
}

// ---------- kernel 1: one workgroup per row ----------

__global__ __launch_bounds__(256) void gem_row_kernel(
    const float* __restrict__ logits, const int* __restrict__ labels,
    float* __restrict__ per_tok, float* __restrict__ vld, int V) {
  const int row  = blockIdx.x;
  const int tid  = threadIdx.x;
  const int w    = tid >> 5;
  const int lane = tid & 31;
  const float* __restrict__ base = logits + (long long)row * V;

  __shared__ float4 stage[WAVES * SLOTS * 32];   // 32 KB ring buffers (512B per slot per wave)
  __shared__ float  m8[WAVES], S8[WAVES], E8[WAVES];

  const int nchunks = V >> 7;                    // 128 floats (512B) per chunk
  int n = (nchunks - w + WAVES - 1) / WAVES;     // this wave handles chunks c = w, w+8, ...
  if (n < 0) n = 0;

  float m = -__builtin_inff(), S = 0.f, E = 0.f;

  // pipeline prologue: issue DEPTH async loads
  const int pre = (n < DEPTH) ? n : DEPTH;
  for (int k = 0; k < pre; ++k) {
    int g = w + WAVES * k;
    unsigned goff = (unsigned)g * 512u + (unsigned)lane * 16u;
    unsigned loff = lds_byte_off(&stage[(w * SLOTS + (k & (SLOTS - 1))) * 32 + lane]);
    async_copy_b128(base, goff, loff);
  }

  for (int i = 0; i < n; ++i) {
    if (i + DEPTH <= n) wait_asynccnt<DEPTH - 1>();  // oldest (chunk i) complete
    else                wait_asynccnt<0>();          // drain phase
    float4 v = stage[(w * SLOTS + (i & (SLOTS - 1))) * 32 + lane];
    int nx = i + DEPTH;
    if (nx < n) {                                    // refill the slot read DEPTH (4) iterations ahead
      int g = w + WAVES * nx;
      unsigned goff = (unsigned)g * 512u + (unsigned)lane * 16u;
      unsigned loff = lds_byte_off(&stage[(w * SLOTS + (nx & (SLOTS - 1))) * 32 + lane]);
      async_copy_b128(base, goff, loff);
    }
    upd(v.x, m, S, E); upd(v.y, m, S, E); upd(v.z, m, S, E); upd(v.w, m, S, E);
  }

  // tail if V is not a multiple of 128 (not the case for V=32000, kept for generality)
  for (int idx = (nchunks << 7) + tid; idx < V; idx += 256)
    upd(base[idx], m, S, E);

  // wave32 butterfly reduce
#pragma unroll
  for (int off = 16; off > 0; off >>= 1) {
    float mm = __shfl_xor(m, off, 32);
    float ss = __shfl_xor(S, off, 32);
    float ee = __shfl_xor(E, off, 32);
    mrg(m, S, E, mm, ss, ee);
  }
  if (lane == 0) { m8[w] = m; S8[w] = S; E8[w] = E; }
  __syncthreads();

  if (tid == 0) {
    float mm = m8[0], SS = S8[0], EE = E8[0];
#pragma unroll
    for (int k = 1; k < WAVES; ++k) mrg(mm, SS, EE, m8[k], S8[k], E8[k]);
    int  lab   = labels[row];
    bool valid = (lab != IGNORE_INDEX);
    float xl   = valid ? base[lab] : 0.f;
    float pt   = EE / SS - xl;          // E_q[x] - x_label
    per_tok[row] = valid ? pt : 0.f;
    vld[row]     = valid ? 1.f : 0.f;
  }
}

// ---------- kernel 2: deterministic final reduction over N rows ----------

__global__ __launch_bounds__(256) void gem_finalize_kernel(
    const float* __restrict__ per_tok, const float* __restrict__ vld,
    float* __restrict__ out, int N) {
  float s = 0.f, c = 0.f;
  for (int i = threadIdx.x; i < N; i += 256) { s += per_tok[i]; c += vld[i]; }
#pragma unroll
  for (int off = 16; off > 0; off >>= 1) {
    s += __shfl_xor(s, off, 32);
    c += __shfl_xor(c, off, 32);
  }
  __shared__ float ss[8], cs[8];
  const int w = threadIdx.x >> 5, lane = threadIdx.x & 31;
  if (lane == 0) { ss[w] = s; cs[w] = c; }
  __syncthreads();
  if (threadIdx.x == 0) {
    float S = 0.f, C = 0.f;
    for (int k = 0; k < 8; ++k) { S += ss[k]; C += cs[k]; }
    out[0] = S / C;
  }
}

extern "C" void kernel_launch(void* const* d_in, const int* in_sizes, int n_in,
                              void* d_out, int out_size, void* d_ws, size_t ws_size,
                              hipStream_t stream) {
  const float* logits = (const float*)d_in[0];
  const int*   labels = (const int*)d_in[1];
  const int N = in_sizes[1];
  const int V = in_sizes[0] / N;

  float* per_tok = (float*)d_ws;      // N floats
  float* vld     = per_tok + N;       // N floats

  gem_row_kernel<<<N, 256, 0, stream>>>(logits, labels, per_tok, vld, V);
  gem_finalize_kernel<<<1, 256, 0, stream>>>(per_tok, vld, (float*)d_out, N);
}